// CustomGINConv_58437325029516
// MI455X (gfx1250) — compile-verified
//
#include <hip/hip_runtime.h>

typedef float v2f __attribute__((ext_vector_type(2)));
typedef float v8f __attribute__((ext_vector_type(8)));

#ifndef __has_builtin
#define __has_builtin(x) 0
#endif
#if defined(__HIP_DEVICE_COMPILE__) && defined(__gfx1250__)
#if !__has_builtin(__builtin_amdgcn_wmma_f32_16x16x4_f32)
#error "wmma_f32_16x16x4_f32 builtin not available on gfx1250 device pass"
#endif
#endif

#define DFEAT 64
#define WSTRIDE 66   // padded LDS row stride (floats): keeps 8B alignment, spreads banks

// ---------------- Kernel 1: out = x  (the (1+eps)*x term, eps=0) -------------
__global__ __launch_bounds__(256) void gin_init_kernel(const float4* __restrict__ x4,
                                                       float4* __restrict__ out4,
                                                       int n4) {
    int i = blockIdx.x * 256 + threadIdx.x;
    if (i < n4) out4[i] = x4[i];
}

// ---------------- Kernel 2: scatter-add x[src] into out[dst] -----------------
// 256 threads = 4 edges/block, 64 threads (one per feature) per edge.
__global__ __launch_bounds__(256) void gin_scatter_kernel(const float* __restrict__ x,
                                                          const long long* __restrict__ edge_index,
                                                          float* __restrict__ agg,
                                                          int n_edges) {
    int e = blockIdx.x * 4 + (threadIdx.x >> 6);
    int f = threadIdx.x & 63;
    if (e < n_edges) {
        long long s = edge_index[e];             // src row
        long long d = edge_index[n_edges + e];   // dst row
        float v = x[(size_t)s * DFEAT + f];
        atomicAdd(&agg[(size_t)d * DFEAT + f], v);   // global_atomic_add_f32
    }
}

// ---------------- Kernel 3: fused 2-layer MLP with f32 WMMA ------------------
// h (= out buffer) -> relu(h@W1+b1) -> (@W2+b2) -> out, in place per row tile.
// Per block: 8 waves, each wave computes a 16-row x 64-col tile.
// Weights live transposed in LDS (Wt[col][k], stride 66) so each B fragment is
// one aligned ds_load_b64; one weight buffer is reused for W1 then W2.
__global__ __launch_bounds__(256) void gin_mlp_kernel(const float* __restrict__ h_in,
                                                      const float* __restrict__ W1,
                                                      const float* __restrict__ b1,
                                                      const float* __restrict__ W2,
                                                      const float* __restrict__ b2,
                                                      float* __restrict__ out,
                                                      int n_tiles) {
    __shared__ float sWt[DFEAT * WSTRIDE];      // ~16.5 KB, transposed weights
    __shared__ float sH[8][16 * WSTRIDE];       // ~33 KB, per-wave relu tile (padded)

    const int tid  = threadIdx.x;
    const int wave = tid >> 5;
    const int lane = tid & 31;

    // Cooperative transposed load: sWt[c][r] = W1[r][c]. Coalesced global reads.
    for (int i = tid; i < DFEAT * DFEAT; i += 256) {
        int r = i >> 6, c = i & 63;
        sWt[c * WSTRIDE + r] = W1[i];
    }
    __syncthreads();

    // Tile assignment; clamp idle tail waves so everyone reaches barriers.
    int tile  = blockIdx.x * 8 + wave;
    bool live = (tile < n_tiles);
    int t     = live ? tile : 0;
    const size_t row0 = (size_t)t * 16;

    const int laneHi = lane >> 4;   // 0: lanes 0-15, 1: lanes 16-31
    const int lm     = lane & 15;

    // ---------------- Layer 1: acc = h_tile @ W1 ----------------
    // A frag (16x4 f32): lanes 0-15 row M=lm hold K=4k+0,4k+1;
    //                    lanes 16-31 row M=lm hold K=4k+2,4k+3. B mirrored.
    v8f acc0 = {}, acc1 = {}, acc2 = {}, acc3 = {};
    const float* hrow = h_in + (row0 + lm) * DFEAT;

    for (int k = 0; k < 16; ++k) {
        const int kb = k * 4 + laneHi * 2;            // even -> 8B aligned
        v2f a = *(const v2f*)&hrow[kb];               // global_load_b64
        v2f b0 = *(const v2f*)&sWt[(0 * 16 + lm) * WSTRIDE + kb];   // ds_load_b64
        v2f b1v = *(const v2f*)&sWt[(1 * 16 + lm) * WSTRIDE + kb];
        v2f b2v = *(const v2f*)&sWt[(2 * 16 + lm) * WSTRIDE + kb];
        v2f b3 = *(const v2f*)&sWt[(3 * 16 + lm) * WSTRIDE + kb];
        acc0 = __builtin_amdgcn_wmma_f32_16x16x4_f32(false, a, false, b0,  (short)0, acc0, false, false);
        acc1 = __builtin_amdgcn_wmma_f32_16x16x4_f32(false, a, false, b1v, (short)0, acc1, false, false);
        acc2 = __builtin_amdgcn_wmma_f32_16x16x4_f32(false, a, false, b2v, (short)0, acc2, false, false);
        acc3 = __builtin_amdgcn_wmma_f32_16x16x4_f32(false, a, false, b3,  (short)0, acc3, false, false);
    }

    // Bias + ReLU, stage in LDS (padded row-major 16x64 tile) for layer-2 A frags.
    // C/D layout: VGPR i -> rows M=i (lanes 0-15) / M=i+8 (lanes 16-31), col = lm + 16n.
    float* myH = &sH[wave][0];
    {
        v8f accs[4] = {acc0, acc1, acc2, acc3};
        #pragma unroll
        for (int n = 0; n < 4; ++n) {
            float bias = b1[n * 16 + lm];
            #pragma unroll
            for (int i = 0; i < 8; ++i) {
                float v = accs[n][i] + bias;
                v = v > 0.0f ? v : 0.0f;
                int row = i + laneHi * 8;
                myH[row * WSTRIDE + n * 16 + lm] = v;   // conflict-free banks
            }
        }
    }
    __syncthreads();   // all waves done reading W1t + sH stores visible

    // Swap in transposed W2, reusing the same LDS buffer.
    for (int i = tid; i < DFEAT * DFEAT; i += 256) {
        int r = i >> 6, c = i & 63;
        sWt[c * WSTRIDE + r] = W2[i];
    }
    __syncthreads();

    // ---------------- Layer 2: acc = relu_tile @ W2 ----------------
    v8f o0 = {}, o1 = {}, o2 = {}, o3 = {};
    for (int k = 0; k < 16; ++k) {
        const int kb = k * 4 + laneHi * 2;
        v2f a = *(const v2f*)&myH[lm * WSTRIDE + kb];  // aligned ds_load_b64, no 16-way conflict
        v2f b0 = *(const v2f*)&sWt[(0 * 16 + lm) * WSTRIDE + kb];
        v2f b1v = *(const v2f*)&sWt[(1 * 16 + lm) * WSTRIDE + kb];
        v2f b2v = *(const v2f*)&sWt[(2 * 16 + lm) * WSTRIDE + kb];
        v2f b3 = *(const v2f*)&sWt[(3 * 16 + lm) * WSTRIDE + kb];
        o0 = __builtin_amdgcn_wmma_f32_16x16x4_f32(false, a, false, b0,  (short)0, o0, false, false);
        o1 = __builtin_amdgcn_wmma_f32_16x16x4_f32(false, a, false, b1v, (short)0, o1, false, false);
        o2 = __builtin_amdgcn_wmma_f32_16x16x4_f32(false, a, false, b2v, (short)0, o2, false, false);
        o3 = __builtin_amdgcn_wmma_f32_16x16x4_f32(false, a, false, b3,  (short)0, o3, false, false);
    }

    // Store with bias. Only live tiles write (in-place over h rows we own).
    if (live) {
        v8f outs[4] = {o0, o1, o2, o3};
        #pragma unroll
        for (int n = 0; n < 4; ++n) {
            float bias = b2[n * 16 + lm];
            #pragma unroll
            for (int i = 0; i < 8; ++i) {
                size_t row = row0 + i + laneHi * 8;
                out[row * DFEAT + n * 16 + lm] = outs[n][i] + bias;
            }
        }
    }
}

// ------------------------------- launcher ------------------------------------
extern "C" void kernel_launch(void* const* d_in, const int* in_sizes, int n_in,
                              void* d_out, int out_size, void* d_ws, size_t ws_size,
                              hipStream_t stream) {
    const float*     x  = (const float*)d_in[0];
    const long long* ei = (const long long*)d_in[1];   // int64 edge_index [2, E]
    const float*     W1 = (const float*)d_in[2];
    const float*     b1 = (const float*)d_in[3];
    const float*     W2 = (const float*)d_in[4];
    const float*     b2 = (const float*)d_in[5];
    float*           out = (float*)d_out;

    const int n_nodes = in_sizes[0] / DFEAT;
    const int n_edges = in_sizes[1] / 2;

    // 1) out = x  (vectorized float4 copy)
    const int n4 = n_nodes * DFEAT / 4;
    gin_init_kernel<<<(n4 + 255) / 256, 256, 0, stream>>>(
        (const float4*)x, (float4*)out, n4);

    // 2) out[dst] += x[src] via global fp32 atomics (L2-resident)
    gin_scatter_kernel<<<(n_edges + 3) / 4, 256, 0, stream>>>(x, ei, out, n_edges);

    // 3) fused MLP in place: out = relu(out@W1+b1)@W2 + b2
    const int n_tiles = (n_nodes + 15) / 16;
    const int blocks  = (n_tiles + 7) / 8;
    gin_mlp_kernel<<<blocks, 256, 0, stream>>>(out, W1, b1, W2, b2, out, n_tiles);
}